// MPNNLayer_55095840473648
// MI455X (gfx1250) — compile-verified
//
#include <hip/hip_runtime.h>

typedef __attribute__((ext_vector_type(2))) float v2f;
typedef __attribute__((ext_vector_type(8))) float v8f;

#define N_NODES 100000
#define N_EDGES 800000
#define HIDDEN  128
#define SCAN_THREADS 1024

// ---------------------------------------------------------------------------
// Kernel 1: fused triple GEMM (unchanged, WMMA f32 16x16x4).
//   Wh  = h @ W_w.T  + W_b   -> ws
//   Wsh = h @ Ws_w.T + Ws_b  -> d_out (initializes the accumulator buffer)
//   Wth = h @ Wt_w.T + Wt_b  -> ws
// ---------------------------------------------------------------------------
__global__ __launch_bounds__(256) void gemm3_wmma_kernel(
    const float* __restrict__ h,
    const float* __restrict__ Ww,  const float* __restrict__ Wb,
    const float* __restrict__ Wsw, const float* __restrict__ Wsb,
    const float* __restrict__ Wtw, const float* __restrict__ Wtb,
    float* __restrict__ outWh, float* __restrict__ outWsh,
    float* __restrict__ outWth)
{
    __shared__ float Ash[16][132];   // +4 pad: bank = (4*m + k) & 63, conflict-free

    const int tid = threadIdx.x;
    const int m0  = blockIdx.x * 16;

    // Cooperative load of the 16x128 fp32 A tile (two float4 per thread).
    {
        const int r = tid >> 4;            // 0..15
        const int c = (tid & 15) * 8;      // 0..120, float4-aligned
        const float4* src = (const float4*)(h + (size_t)(m0 + r) * HIDDEN + c);
        float4 v0 = src[0];
        float4 v1 = src[1];
        *(float4*)&Ash[r][c]     = v0;
        *(float4*)&Ash[r][c + 4] = v1;
    }
    __syncthreads();

    const int wave = tid >> 5;             // 0..7 : 16-col tile
    const int lane = tid & 31;
    const int lg   = lane >> 4;            // half-wave selects K pair {0,1} vs {2,3}
    const int nl   = lane & 15;
    const int n    = wave * 16 + nl;       // output column

    const float* wp  = Ww  + (size_t)n * HIDDEN;
    const float* wsp = Wsw + (size_t)n * HIDDEN;
    const float* wtp = Wtw + (size_t)n * HIDDEN;

    v8f accW = {};
    v8f accS = {};
    v8f accT = {};

    #pragma unroll 4
    for (int kc = 0; kc < 32; ++kc) {
        const int k0 = kc * 4 + 2 * lg;
        v2f a  = *(const v2f*)&Ash[nl][k0];
        v2f bW = *(const v2f*)(wp  + k0);
        v2f bS = *(const v2f*)(wsp + k0);
        v2f bT = *(const v2f*)(wtp + k0);
        accW = __builtin_amdgcn_wmma_f32_16x16x4_f32(false, a, false, bW,
                                                     (short)0, accW, false, false);
        accS = __builtin_amdgcn_wmma_f32_16x16x4_f32(false, a, false, bS,
                                                     (short)0, accS, false, false);
        accT = __builtin_amdgcn_wmma_f32_16x16x4_f32(false, a, false, bT,
                                                     (short)0, accT, false, false);
    }

    const float bw = Wb[n];
    const float bs = Wsb[n];
    const float bt = Wtb[n];
    const int mbase = m0 + 8 * lg;
    #pragma unroll
    for (int r = 0; r < 8; ++r) {
        const size_t idx = (size_t)(mbase + r) * HIDDEN + n;
        outWh[idx]  = accW[r] + bw;
        outWsh[idx] = accS[r] + bs;
        outWth[idx] = accT[r] + bt;
    }
}

// ---------------------------------------------------------------------------
// CSR build step 1: zero the degree counters (2N ints).
// ---------------------------------------------------------------------------
__global__ __launch_bounds__(256) void zero_kernel(int* __restrict__ p, int n)
{
    const int i = (int)(blockIdx.x * blockDim.x + threadIdx.x);
    if (i < n) p[i] = 0;
}

// ---------------------------------------------------------------------------
// CSR build step 2: count out-degree (by src) and in-degree (by dst).
// ---------------------------------------------------------------------------
__global__ __launch_bounds__(256) void count_kernel(
    const int* __restrict__ esrc, const int* __restrict__ edst,
    int* __restrict__ cnt_f, int* __restrict__ cnt_b)
{
    const int e = (int)(blockIdx.x * blockDim.x + threadIdx.x);
    if (e >= N_EDGES) return;
    atomicAdd(&cnt_f[esrc[e]], 1);
    atomicAdd(&cnt_b[edst[e]], 1);
}

// ---------------------------------------------------------------------------
// CSR build step 3: exclusive prefix sum of both degree arrays.
// Single 1024-thread block; per-thread serial chunk (~98 nodes) + LDS
// Hillis-Steele scan of the 1024 partials. Also seeds the fill cursors.
// ---------------------------------------------------------------------------
__global__ __launch_bounds__(SCAN_THREADS) void scan_kernel(
    const int* __restrict__ cnt_f, const int* __restrict__ cnt_b,
    int* __restrict__ off_f, int* __restrict__ off_b,
    int* __restrict__ cur_f, int* __restrict__ cur_b)
{
    __shared__ int part[SCAN_THREADS];
    const int t = (int)threadIdx.x;
    const int chunk = (N_NODES + SCAN_THREADS - 1) / SCAN_THREADS;   // 98

    for (int pass = 0; pass < 2; ++pass) {
        const int* cnt = pass ? cnt_b : cnt_f;
        int* off = pass ? off_b : off_f;
        int* cur = pass ? cur_b : cur_f;

        const int lo = t * chunk;
        int hi = lo + chunk;
        if (hi > N_NODES) hi = N_NODES;

        int sum = 0;
        for (int i = lo; i < hi; ++i) sum += cnt[i];
        part[t] = sum;
        __syncthreads();

        for (int d = 1; d < SCAN_THREADS; d <<= 1) {
            const int add = (t >= d) ? part[t - d] : 0;
            __syncthreads();
            part[t] += add;
            __syncthreads();
        }

        int run = (t == 0) ? 0 : part[t - 1];
        for (int i = lo; i < hi; ++i) {
            off[i] = run;
            cur[i] = run;
            run += cnt[i];
        }
        if (t == SCAN_THREADS - 1) off[N_NODES] = part[SCAN_THREADS - 1];
        __syncthreads();
    }
}

// ---------------------------------------------------------------------------
// CSR build step 4: bin fill.
//   bin_f (indexed by src) holds dst ids; bin_b (indexed by dst) holds src ids.
// ---------------------------------------------------------------------------
__global__ __launch_bounds__(256) void fill_kernel(
    const int* __restrict__ esrc, const int* __restrict__ edst,
    int* __restrict__ cur_f, int* __restrict__ cur_b,
    int* __restrict__ bin_f, int* __restrict__ bin_b)
{
    const int e = (int)(blockIdx.x * blockDim.x + threadIdx.x);
    if (e >= N_EDGES) return;
    const int s = esrc[e];
    const int d = edst[e];
    const int p = atomicAdd(&cur_f[s], 1);
    bin_f[p] = d;
    const int q = atomicAdd(&cur_b[d], 1);
    bin_b[q] = s;
}

// ---------------------------------------------------------------------------
// Kernel 5: wave-per-node gather aggregation + bias'd self term + ReLU.
// Each lane owns 4 of the 128 channels in registers; neighbor rows are
// float4-gathered (Wh/Wth together = 102 MB -> L2-resident). Single
// coalesced store, no float atomics anywhere.
// ---------------------------------------------------------------------------
__global__ __launch_bounds__(256) void aggregate_kernel(
    const float* __restrict__ Wh, const float* __restrict__ Wth,
    const int* __restrict__ off_f, const int* __restrict__ bin_f,
    const int* __restrict__ off_b, const int* __restrict__ bin_b,
    float* __restrict__ out)
{
    const int node = (int)((blockIdx.x * blockDim.x + threadIdx.x) >> 5);
    if (node >= N_NODES) return;
    const int lane = (int)(threadIdx.x & 31);
    const int c    = lane * 4;

    // Self term Wsh (with bias) was written into d_out by the GEMM.
    float4 acc = *(const float4*)(out + (size_t)node * HIDDEN + c);

    // agg_fwd[node] = sum over out-edges (node -> d) of Wh[d]
    const int f0 = off_f[node], f1 = off_f[node + 1];
    for (int j = f0; j < f1; ++j) {
        const int nb = bin_f[j];
        const float4 v = *(const float4*)(Wh + (size_t)nb * HIDDEN + c);
        acc.x += v.x; acc.y += v.y; acc.z += v.z; acc.w += v.w;
    }

    // agg_bwd[node] = sum over in-edges (s -> node) of Wth[s]
    const int b0 = off_b[node], b1 = off_b[node + 1];
    for (int j = b0; j < b1; ++j) {
        const int nb = bin_b[j];
        const float4 v = *(const float4*)(Wth + (size_t)nb * HIDDEN + c);
        acc.x += v.x; acc.y += v.y; acc.z += v.z; acc.w += v.w;
    }

    acc.x = fmaxf(acc.x, 0.0f);
    acc.y = fmaxf(acc.y, 0.0f);
    acc.z = fmaxf(acc.z, 0.0f);
    acc.w = fmaxf(acc.w, 0.0f);
    *(float4*)(out + (size_t)node * HIDDEN + c) = acc;
}

extern "C" void kernel_launch(void* const* d_in, const int* in_sizes, int n_in,
                              void* d_out, int out_size, void* d_ws, size_t ws_size,
                              hipStream_t stream)
{
    const float* h_n  = (const float*)d_in[0];
    const int*   esrc = (const int*)  d_in[1];
    const int*   edst = (const int*)  d_in[2];
    const float* W_w  = (const float*)d_in[3];
    const float* W_b  = (const float*)d_in[4];
    const float* Ws_w = (const float*)d_in[5];
    const float* Ws_b = (const float*)d_in[6];
    const float* Wt_w = (const float*)d_in[7];
    const float* Wt_b = (const float*)d_in[8];

    float* out = (float*)d_out;

    // Workspace layout
    float* Wh  = (float*)d_ws;                          // [N, H]
    float* Wth = Wh + (size_t)N_NODES * HIDDEN;         // [N, H]
    int*   ip    = (int*)(Wth + (size_t)N_NODES * HIDDEN);
    int*   cnt_f = ip;                 ip += N_NODES;
    int*   cnt_b = ip;                 ip += N_NODES;
    int*   off_f = ip;                 ip += N_NODES + 1;
    int*   off_b = ip;                 ip += N_NODES + 1;
    int*   cur_f = ip;                 ip += N_NODES;
    int*   cur_b = ip;                 ip += N_NODES;
    int*   bin_f = ip;                 ip += N_EDGES;
    int*   bin_b = ip;                 ip += N_EDGES;

    // 1) Triple GEMM (WMMA). Wsh (+bias) goes straight into d_out.
    const int mtiles = N_NODES / 16;                    // 6250, exact
    gemm3_wmma_kernel<<<mtiles, 256, 0, stream>>>(
        h_n, W_w, W_b, Ws_w, Ws_b, Wt_w, Wt_b, Wh, out, Wth);

    // 2) CSR build: zero counters, count degrees, scan, fill bins.
    zero_kernel<<<(2 * N_NODES + 255) / 256, 256, 0, stream>>>(cnt_f, 2 * N_NODES);
    const int eblocks = (N_EDGES + 255) / 256;          // 3125
    count_kernel<<<eblocks, 256, 0, stream>>>(esrc, edst, cnt_f, cnt_b);
    scan_kernel<<<1, SCAN_THREADS, 0, stream>>>(cnt_f, cnt_b,
                                                off_f, off_b, cur_f, cur_b);
    fill_kernel<<<eblocks, 256, 0, stream>>>(esrc, edst, cur_f, cur_b,
                                             bin_f, bin_b);

    // 3) Wave-per-node gather + ReLU (no float atomics).
    const int ablocks = (N_NODES * 32) / 256;           // 12500, exact
    aggregate_kernel<<<ablocks, 256, 0, stream>>>(Wh, Wth, off_f, bin_f,
                                                  off_b, bin_b, out);
}